// JaggedMaxModule_49314814492717
// MI455X (gfx1250) — compile-verified
//
#include <hip/hip_runtime.h>

// Jagged segment-max, MI455X (gfx1250, wave32).
// One wave per segment; 32 rows x 16 features per chunk-iteration.
// Scalar (s_load) segment bounds, uniform scalar loop control,
// non-temporal 128-bit loads (1 GiB single-use stream), full-chunk
// global_prefetch_b8 lookahead, wave32 shfl_xor tree reduction, NT stores.

typedef float v4f __attribute__((ext_vector_type(4)));

#define SEG_D 16
#define WAVES_PER_BLOCK 8

__device__ __forceinline__ v4f vmax4(v4f a, v4f b) {
    v4f r;
    r.x = fmaxf(a.x, b.x);
    r.y = fmaxf(a.y, b.y);
    r.z = fmaxf(a.z, b.z);
    r.w = fmaxf(a.w, b.w);
    return r;
}

__global__ void __launch_bounds__(WAVES_PER_BLOCK * 32)
jagged_seg_max_kernel(const float* __restrict__ values,
                      const long long* __restrict__ prefix,
                      float* __restrict__ out,
                      int num_segments) {
    const int lane = threadIdx.x & 31;

    // Wave-uniform segment id -> SGPR, so bounds become s_load / SALU.
    int seg = blockIdx.x * WAVES_PER_BLOCK + (threadIdx.x >> 5);
    seg = __builtin_amdgcn_readfirstlane(seg);
    if (seg >= num_segments) return;   // uniform s_cbranch exit

    const long long beg   = prefix[seg];
    const long long end   = prefix[seg + 1];
    const long long nrows = end - beg;

    // Lane mapping: 8 rows per 32-row chunk slice, 4 lanes per row (float4).
    const int rlane   = lane >> 2;               // row offset 0..7
    const int cq      = (lane & 3) << 2;         // feature quad start
    const int laneoff = rlane * (SEG_D * 4) + cq * 4;  // byte offset in chunk

    const float NEG_INF = -__builtin_huge_valf();
    v4f acc = { NEG_INF, NEG_INF, NEG_INF, NEG_INF };

    const char* segbase = (const char*)(values + (size_t)beg * SEG_D);

    // Main loop: uniform trip count, 32 rows (2 KB) per chunk,
    // 4 independent NT B128 loads in flight per lane.
    const long long nfull = nrows >> 5;          // full 32-row chunks
    const char* cp = segbase;                    // chunk base
    for (long long i = 0; i < nfull; ++i) {
        const v4f* p = (const v4f*)(cp + laneoff);
        v4f a0 = __builtin_nontemporal_load(p);        //  rows +0..7
        v4f a1 = __builtin_nontemporal_load(p + 32);   //  rows +8..15
        v4f a2 = __builtin_nontemporal_load(p + 64);   //  rows +16..23
        v4f a3 = __builtin_nontemporal_load(p + 96);   //  rows +24..31
        // Prefetch the entire next 2 KB chunk: 32 lanes x 64B stride.
        __builtin_prefetch(cp + 2048 + lane * 64, 0, 1);
        acc = vmax4(acc, vmax4(vmax4(a0, a1), vmax4(a2, a3)));
        cp += 2048;
    }

    // Remainder: up to 31 rows, 8 per iteration with per-lane bound check.
    long long r = (nfull << 5) + rlane;
    const v4f* p = (const v4f*)(cp + laneoff);
    for (; r < nrows; r += 8) {
        v4f a = __builtin_nontemporal_load(p);
        acc = vmax4(acc, a);
        p += 32;
    }

    // Wave32 tree reduction across row-lanes (same feature quad: lane & 3).
#pragma unroll
    for (int m = 4; m < 32; m <<= 1) {
        acc.x = fmaxf(acc.x, __shfl_xor(acc.x, m, 32));
        acc.y = fmaxf(acc.y, __shfl_xor(acc.y, m, 32));
        acc.z = fmaxf(acc.z, __shfl_xor(acc.z, m, 32));
        acc.w = fmaxf(acc.w, __shfl_xor(acc.w, m, 32));
    }

    // Lanes 0..3 hold quads 0..3 -> one NT B128 store each (64B per segment).
    if (lane < 4) {
        float* o = out + (size_t)seg * SEG_D + (lane << 2);
        __builtin_nontemporal_store(acc, reinterpret_cast<v4f*>(o));
    }
}

extern "C" void kernel_launch(void* const* d_in, const int* in_sizes, int n_in,
                              void* d_out, int out_size, void* d_ws, size_t ws_size,
                              hipStream_t stream) {
    const float*     values = (const float*)d_in[0];
    const long long* prefix = (const long long*)d_in[1];  // int64 prefix_sum
    float*           out    = (float*)d_out;

    const int num_segments = in_sizes[1] - 1;  // S = len(prefix_sum) - 1
    const int blocks = (num_segments + WAVES_PER_BLOCK - 1) / WAVES_PER_BLOCK;

    jagged_seg_max_kernel<<<blocks, WAVES_PER_BLOCK * 32, 0, stream>>>(
        values, prefix, out, num_segments);
}